// GCNLinkPrediction_50766513439535
// MI455X (gfx1250) — compile-verified
//
#include <hip/hip_runtime.h>
#include <hip/hip_bf16.h>

typedef float __attribute__((ext_vector_type(2))) v2f;
typedef float __attribute__((ext_vector_type(8))) v8f;

// ---------------------------------------------------------------------------
// Degree / normalization kernels
// ---------------------------------------------------------------------------
__global__ void deg_init_kernel(float* __restrict__ deg, int n) {
    int i = blockIdx.x * blockDim.x + threadIdx.x;
    if (i < n) deg[i] = 1.0f;   // self loop
}

__global__ void deg_edges_kernel(const int* __restrict__ dst,
                                 float* __restrict__ deg, int E) {
    int e = blockIdx.x * blockDim.x + threadIdx.x;
    if (e < E) unsafeAtomicAdd(&deg[dst[e]], 1.0f);
}

__global__ void deg_to_dinv_kernel(float* __restrict__ deg, int n) {
    int i = blockIdx.x * blockDim.x + threadIdx.x;
    if (i < n) deg[i] = rsqrtf(deg[i]);
}

// ---------------------------------------------------------------------------
// Dense GEMM: Y[M,ND] = relu?(X[M,KD]) @ W[KD,ND], scaled per-row by dinv[m],
// via V_WMMA_F32_16X16X4_F32. One workgroup = 8 waves = 128 rows; W in LDS.
// RELU applies the previous layer's activation on the fly (saves a full pass).
// The dinv[m] row scale pre-applies one normalization factor so the scatter
// kernel only needs dinv[dst].
// ---------------------------------------------------------------------------
template <int KD, int ND, bool RELU>
__global__ __launch_bounds__(256)
void gemm_wmma_f32(const float* __restrict__ X, const float* __restrict__ W,
                   float* __restrict__ Y, const float* __restrict__ dinv, int M)
{
    constexpr int NT = ND / 16;          // number of 16-col output tiles per wave
    __shared__ float lw[KD * ND];        // staged weight matrix (<= 64 KB)

    const int tid = threadIdx.x;
    constexpr int TOT4 = (KD * ND) / 4;
    for (int i = tid; i < TOT4; i += 256)
        ((float4*)lw)[i] = ((const float4*)W)[i];
    __syncthreads();

    const int lane   = tid & 31;
    const int waveId = tid >> 5;
    const int row0   = blockIdx.x * 128 + waveId * 16;
    const int r      = lane & 15;          // row-in-tile (A) / col-in-tile (B,D)
    const int khalf  = (lane >> 4) << 1;   // 0 for lanes 0-15, 2 for lanes 16-31

    int arow = row0 + r;
    if (arow >= M) arow = M - 1;           // clamp: keeps EXEC all-ones for WMMA
    const float* xrow = X + (long)arow * KD;

    v8f acc[NT];
#pragma unroll
    for (int t = 0; t < NT; ++t)
        acc[t] = (v8f){0.f, 0.f, 0.f, 0.f, 0.f, 0.f, 0.f, 0.f};

    for (int k = 0; k < KD; k += 4) {
        v2f a;                              // A 16x4: VGPR0=K(khalf), VGPR1=K(khalf+1)
        float a0 = xrow[k + khalf];
        float a1 = xrow[k + khalf + 1];
        if (RELU) { a0 = fmaxf(a0, 0.f); a1 = fmaxf(a1, 0.f); }
        a.x = a0; a.y = a1;
#pragma unroll
        for (int t = 0; t < NT; ++t) {
            const int col = t * 16 + r;
            v2f b;                          // B 4x16: rows striped across lanes
            b.x = lw[(k + khalf) * ND + col];
            b.y = lw[(k + khalf + 1) * ND + col];
            acc[t] = __builtin_amdgcn_wmma_f32_16x16x4_f32(
                false, a, false, b, (short)0, acc[t], false, false);
        }
    }

    // D layout: VGPR v -> M = row0 + v (lanes 0-15) / row0 + v + 8 (lanes 16-31)
    const int mhi = (lane >> 4) << 3;

    // per-row normalization factors (8 rows this lane-half contributes to)
    float dvm[8];
#pragma unroll
    for (int v = 0; v < 8; ++v) {
        int m = row0 + v + mhi;
        if (m >= M) m = M - 1;
        dvm[v] = dinv[m];
    }

    if (row0 + 16 <= M) {
        // full tile: straight-line stores, 32-bit offsets off SGPR base
#pragma unroll
        for (int t = 0; t < NT; ++t) {
            const int col = t * 16 + r;
#pragma unroll
            for (int v = 0; v < 8; ++v) {
                const int m = row0 + v + mhi;
                Y[m * ND + col] = acc[t][v] * dvm[v];
            }
        }
    } else {
        // partial tile at the tail of the matrix
#pragma unroll
        for (int t = 0; t < NT; ++t) {
            const int col = t * 16 + r;
#pragma unroll
            for (int v = 0; v < 8; ++v) {
                const int m = row0 + v + mhi;
                if (m < M) Y[m * ND + col] = acc[t][v] * dvm[v];
            }
        }
    }
}

// ---------------------------------------------------------------------------
// Aggregation. lin already carries one dinv factor (applied at GEMM store):
//   init:  out[i] = lin[i]*dinv[i] + bias        (self loop: dinv[i]^2 total)
//   edges: out[dst] += lin[src]*dinv[dst]        (dinv[src]*dinv[dst] total)
// ---------------------------------------------------------------------------
template <int D>
__global__ void agg_init_kernel(const float* __restrict__ lin,
                                const float* __restrict__ dinv,
                                const float* __restrict__ bias,
                                float* __restrict__ out, int Nn)
{
    constexpr int C4 = D / 4;
    long t = (long)blockIdx.x * blockDim.x + threadIdx.x;
    if (t >= (long)Nn * C4) return;
    int i = (int)(t / C4);
    int c = (int)(t % C4) * 4;
    float w = dinv[i];
    float4 l = *(const float4*)(lin + (long)i * D + c);
    float4 o;
    o.x = l.x * w + bias[c + 0];
    o.y = l.y * w + bias[c + 1];
    o.z = l.z * w + bias[c + 2];
    o.w = l.w * w + bias[c + 3];
    *(float4*)(out + (long)i * D + c) = o;
}

template <int D>
__global__ void agg_edges_kernel(const float* __restrict__ lin,
                                 const float* __restrict__ dinv,
                                 const int* __restrict__ src,
                                 const int* __restrict__ dst,
                                 float* __restrict__ out, int E)
{
    constexpr int C4 = D / 4;               // float4 chunks per row
    long t = (long)blockIdx.x * blockDim.x + threadIdx.x;
    if (t >= (long)E * C4) return;
    int e = (int)(t / C4);                  // whole wave shares one edge (D=128)
    int c = (int)(t % C4) * 4;
    int s = src[e];
    int d = dst[e];
    float w = dinv[d];                      // dinv[src] pre-applied in lin
    float4 l = *(const float4*)(lin + (long)s * D + c);
    float* o = out + (long)d * D + c;
    unsafeAtomicAdd(o + 0, l.x * w);
    unsafeAtomicAdd(o + 1, l.y * w);
    unsafeAtomicAdd(o + 2, l.z * w);
    unsafeAtomicAdd(o + 3, l.w * w);
}

// ---------------------------------------------------------------------------
// Decode: scores[i] = dot(z[a], z[b]) over 64 dims
// ---------------------------------------------------------------------------
__global__ void decode_kernel(const float* __restrict__ z,
                              const int* __restrict__ eli,
                              float* __restrict__ out, int L)
{
    int i = blockIdx.x * blockDim.x + threadIdx.x;
    if (i >= L) return;
    int a = eli[i];
    int b = eli[L + i];
    const float4* za = (const float4*)(z + (long)a * 64);
    const float4* zb = (const float4*)(z + (long)b * 64);
    float s = 0.f;
#pragma unroll
    for (int j = 0; j < 16; ++j) {
        float4 x = za[j], y = zb[j];
        s += x.x * y.x + x.y * y.y + x.z * y.z + x.w * y.w;
    }
    out[i] = s;
}

// ---------------------------------------------------------------------------
// Launch
// ---------------------------------------------------------------------------
static inline int cdiv(long a, long b) { return (int)((a + b - 1) / b); }

extern "C" void kernel_launch(void* const* d_in, const int* in_sizes, int n_in,
                              void* d_out, int out_size, void* d_ws, size_t ws_size,
                              hipStream_t stream)
{
    const float* x   = (const float*)d_in[0];
    const int*   ei  = (const int*)  d_in[1];
    const int*   eli = (const int*)  d_in[2];
    const float* W1  = (const float*)d_in[3];
    const float* b1  = (const float*)d_in[4];
    const float* W2  = (const float*)d_in[5];
    const float* b2  = (const float*)d_in[6];
    const float* W3  = (const float*)d_in[7];
    const float* b3  = (const float*)d_in[8];
    float* scores    = (float*)d_out;

    const int N = in_sizes[0] / 128;    // 50000 nodes
    const int E = in_sizes[1] / 2;      // 600000 edges
    const int L = in_sizes[2] / 2;      // 200000 label pairs

    float* dinv = (float*)d_ws;
    float* bufA = dinv + N;                  // N x 128 (linear output, dinv-scaled)
    float* bufB = bufA + (size_t)N * 128;    // N x 128 (aggregated, pre-activation)

    const int* src = ei;
    const int* dst = ei + E;

    const int T = 256;

    // --- symmetric normalization ---
    deg_init_kernel<<<cdiv(N, T), T, 0, stream>>>(dinv, N);
    deg_edges_kernel<<<cdiv(E, T), T, 0, stream>>>(dst, dinv, E);
    deg_to_dinv_kernel<<<cdiv(N, T), T, 0, stream>>>(dinv, N);

    // --- layer 1: bufB = agg(x @ W1) + b1   (relu deferred into next gemm) ---
    gemm_wmma_f32<128, 128, false><<<cdiv(N, 128), 256, 0, stream>>>(x, W1, bufA, dinv, N);
    agg_init_kernel<128><<<cdiv((long)N * 32, T), T, 0, stream>>>(bufA, dinv, b1, bufB, N);
    agg_edges_kernel<128><<<cdiv((long)E * 32, T), T, 0, stream>>>(bufA, dinv, src, dst, bufB, E);

    // --- layer 2: bufB = agg(relu(bufB) @ W2) + b2 ---
    gemm_wmma_f32<128, 128, true><<<cdiv(N, 128), 256, 0, stream>>>(bufB, W2, bufA, dinv, N);
    agg_init_kernel<128><<<cdiv((long)N * 32, T), T, 0, stream>>>(bufA, dinv, b2, bufB, N);
    agg_edges_kernel<128><<<cdiv((long)E * 32, T), T, 0, stream>>>(bufA, dinv, src, dst, bufB, E);

    // --- layer 3: z = agg(relu(bufB) @ W3) + b3 (no output relu) ---
    gemm_wmma_f32<128, 64, true><<<cdiv(N, 128), 256, 0, stream>>>(bufB, W3, bufA, dinv, N);
    agg_init_kernel<64><<<cdiv((long)N * 16, T), T, 0, stream>>>(bufA, dinv, b3, bufB, N);
    agg_edges_kernel<64><<<cdiv((long)E * 16, T), T, 0, stream>>>(bufA, dinv, src, dst, bufB, E);

    // --- decode ---
    decode_kernel<<<cdiv(L, T), T, 0, stream>>>(bufB, eli, scores, L);
}